// PointTokenizer_49478023250593
// MI455X (gfx1250) — compile-verified
//
#include <hip/hip_runtime.h>
#include <math.h>

#define NUM_B 8
#define NUM_N 8192
#define NUM_G 256
#define GK    32
#define C1    128
#define C2    256
#define C3    512
#define CT    384
#define TOK_TOTAL (NUM_B * NUM_G * CT)

typedef __attribute__((ext_vector_type(2))) float v2f;
typedef __attribute__((ext_vector_type(8))) float v8f;

// ---------------------------------------------------------------------------
// FP32 WMMA 16x16x4 (exact fp32 GEMM on the CDNA5 matrix pipe)
// ---------------------------------------------------------------------------
__device__ __forceinline__ v8f wmma4(v2f a, v2f b, v8f c) {
  // 8 args: (neg_a, A, neg_b, B, c_mod, C, reuse_a, reuse_b)
  return __builtin_amdgcn_wmma_f32_16x16x4_f32(false, a, false, b, (short)0, c,
                                               false, false);
}

// 2M x 2N register-blocked microkernel: 4 loads -> 4 WMMAs per k-step.
// W row-major (global, ldw), B column-major in LDS ([n][k], ld = LDB), N = 32.
template <int KDIM, int LDB>
__device__ __forceinline__ void gemm_2x2(const float* __restrict__ W, int ldw,
                                         const float* B, int m0, int laneN,
                                         int laneHalf, v8f& acc00, v8f& acc01,
                                         v8f& acc10, v8f& acc11) {
  const float* wrow0 = W + (size_t)(m0 + laneN) * (size_t)ldw + 2 * laneHalf;
  const float* wrow1 = wrow0 + (size_t)16 * (size_t)ldw;
  const float* bcol0 = B + laneN * LDB + 2 * laneHalf;
  const float* bcol1 = B + (16 + laneN) * LDB + 2 * laneHalf;
#pragma unroll 4
  for (int kk = 0; kk < KDIM; kk += 4) {
    v2f a0 = *(const v2f*)(wrow0 + kk);
    v2f a1 = *(const v2f*)(wrow1 + kk);
    v2f b0 = *(const v2f*)(bcol0 + kk);
    v2f b1 = *(const v2f*)(bcol1 + kk);
    acc00 = wmma4(a0, b0, acc00);
    acc01 = wmma4(a0, b1, acc01);
    acc10 = wmma4(a1, b0, acc10);
    acc11 = wmma4(a1, b1, acc11);
  }
}

// 1M x 2N microkernel (A fragment shared across both N tiles).
template <int KDIM, int LDB>
__device__ __forceinline__ void gemm_1x2(const float* __restrict__ W, int ldw,
                                         const float* B, int m0, int laneN,
                                         int laneHalf, v8f& acc0, v8f& acc1) {
  const float* wrow  = W + (size_t)(m0 + laneN) * (size_t)ldw + 2 * laneHalf;
  const float* bcol0 = B + laneN * LDB + 2 * laneHalf;
  const float* bcol1 = B + (16 + laneN) * LDB + 2 * laneHalf;
#pragma unroll 4
  for (int kk = 0; kk < KDIM; kk += 4) {
    v2f a  = *(const v2f*)(wrow + kk);
    v2f b0 = *(const v2f*)(bcol0 + kk);
    v2f b1 = *(const v2f*)(bcol1 + kk);
    acc0 = wmma4(a, b0, acc0);
    acc1 = wmma4(a, b1, acc1);
  }
}

// D-tile store with per-channel epilogue op(val, m). D layout: lane holds
// rows m0 + 8*laneHalf + r, column n0 + laneN (column-major dst, ld = ldd).
template <typename F>
__device__ __forceinline__ void store_tile(float* dst, int ldd, const v8f& acc,
                                           int m0, int n0, int laneN,
                                           int laneHalf, F op) {
  int n = n0 + laneN;
#pragma unroll
  for (int rr = 0; rr < 8; ++rr) {
    int m = m0 + 8 * laneHalf + rr;
    dst[n * ldd + m] = op(acc[rr], m);
  }
}

// ---------------------------------------------------------------------------
// Kernel 1: farthest point sampling (1 block per batch, points in registers)
// writes centers (B,G,3) directly into d_out tail
// ---------------------------------------------------------------------------
__global__ __launch_bounds__(1024) void fps_kernel(
    const float* __restrict__ points, float* __restrict__ centers) {
  const int b    = blockIdx.x;
  const int tid  = threadIdx.x;
  const int lane = tid & 31;
  const int wv   = tid >> 5;
  const float* base = points + (size_t)b * NUM_N * 3;

  __shared__ float sLast[3];
  __shared__ int   sWin;
  __shared__ float sWv[32];
  __shared__ int   sWi[32];

  float px[8], py[8], pz[8], dd[8];
#pragma unroll
  for (int i = 0; i < 8; ++i) {
    int p = tid * 8 + i;
    px[i] = base[p * 3 + 0];
    py[i] = base[p * 3 + 1];
    pz[i] = base[p * 3 + 2];
    dd[i] = 1e10f;
  }
  if (tid == 0) {
    sLast[0] = base[0]; sLast[1] = base[1]; sLast[2] = base[2];
    float* co = centers + (size_t)b * NUM_G * 3;
    co[0] = base[0]; co[1] = base[1]; co[2] = base[2];  // fps_idx[0] = 0
  }
  __syncthreads();

  for (int g = 1; g < NUM_G; ++g) {
    float lx = sLast[0], ly = sLast[1], lz = sLast[2];
    float bv = -1.f;
    int   bi = 0;
#pragma unroll
    for (int i = 0; i < 8; ++i) {
      float dx = px[i] - lx, dy = py[i] - ly, dz = pz[i] - lz;
      float d  = dx * dx + dy * dy + dz * dz;
      dd[i]    = fminf(dd[i], d);
      if (dd[i] > bv) { bv = dd[i]; bi = tid * 8 + i; }  // first-index tie pref
    }
#pragma unroll
    for (int off = 16; off > 0; off >>= 1) {  // wave32 argmax
      float ov = __shfl_xor(bv, off, 32);
      int   oi = __shfl_xor(bi, off, 32);
      if (ov > bv || (ov == bv && oi < bi)) { bv = ov; bi = oi; }
    }
    if (lane == 0) { sWv[wv] = bv; sWi[wv] = bi; }
    __syncthreads();
    if (tid < 32) {  // 32 waves -> final reduce in wave 0
      float v  = sWv[tid];
      int   ii = sWi[tid];
#pragma unroll
      for (int off = 16; off > 0; off >>= 1) {
        float ov = __shfl_xor(v, off, 32);
        int   oi = __shfl_xor(ii, off, 32);
        if (ov > v || (ov == v && oi < ii)) { v = ov; ii = oi; }
      }
      if (tid == 0) sWin = ii;
    }
    __syncthreads();
    int wi = sWin;
    if (tid == (wi >> 3)) {  // owner thread broadcasts coords, writes center
      int i = wi & 7;
      sLast[0] = px[i]; sLast[1] = py[i]; sLast[2] = pz[i];
      float* co = centers + ((size_t)b * NUM_G + g) * 3;
      co[0] = px[i]; co[1] = py[i]; co[2] = pz[i];
    }
    __syncthreads();
  }
}

// ---------------------------------------------------------------------------
// Kernel 2: KNN (1 block per center). Distances in LDS, 32 argmin rounds.
// ---------------------------------------------------------------------------
__global__ __launch_bounds__(256) void knn_kernel(
    const float* __restrict__ points, const float* __restrict__ centers,
    int* __restrict__ knnIdx) {
  const int gg   = blockIdx.x;  // b*NUM_G + g
  const int b    = gg / NUM_G;
  const int tid  = threadIdx.x;
  const int lane = tid & 31;
  const int wv   = tid >> 5;
  const float* base = points + (size_t)b * NUM_N * 3;

  __shared__ float sd[NUM_N];
  __shared__ float sWv[8];
  __shared__ int   sWi[8];

  float cx = centers[(size_t)gg * 3 + 0];
  float cy = centers[(size_t)gg * 3 + 1];
  float cz = centers[(size_t)gg * 3 + 2];
  float c2 = cx * cx + cy * cy + cz * cz;

  for (int i = 0; i < NUM_N / 256; ++i) {  // coalesced over lanes
    int   p  = i * 256 + tid;
    float xx = base[p * 3 + 0], yy = base[p * 3 + 1], zz = base[p * 3 + 2];
    float x2 = xx * xx + yy * yy + zz * zz;
    float dt = cx * xx + cy * yy + cz * zz;
    sd[p]    = c2 + x2 - 2.f * dt;  // reference distance formula
  }
  __syncthreads();

  for (int r = 0; r < GK; ++r) {
    float bv = 3.0e38f;
    int   bi = NUM_N;
    for (int i = 0; i < NUM_N / 256; ++i) {
      int   p = i * 256 + tid;
      float v = sd[p];
      if (v < bv || (v == bv && p < bi)) { bv = v; bi = p; }
    }
#pragma unroll
    for (int off = 16; off > 0; off >>= 1) {
      float ov = __shfl_xor(bv, off, 32);
      int   oi = __shfl_xor(bi, off, 32);
      if (ov < bv || (ov == bv && oi < bi)) { bv = ov; bi = oi; }
    }
    if (lane == 0) { sWv[wv] = bv; sWi[wv] = bi; }
    __syncthreads();
    if (tid == 0) {
      float v  = sWv[0];
      int   ii = sWi[0];
      for (int w = 1; w < 8; ++w)
        if (sWv[w] < v || (sWv[w] == v && sWi[w] < ii)) { v = sWv[w]; ii = sWi[w]; }
      knnIdx[(size_t)gg * GK + r] = ii;
      sd[ii] = 3.4e38f;  // mask out
    }
    __syncthreads();
  }
}

// ---------------------------------------------------------------------------
// Kernel 3: encoder. One block (8 waves) per group. All GEMMs via
// v_wmma_f32_16x16x4_f32 with 2Mx2N register blocking (4 WMMA per 4 loads);
// activations column-major in LDS (padded strides).
// ---------------------------------------------------------------------------
__global__ __launch_bounds__(256) void encoder_kernel(
    const float* __restrict__ points, const int* __restrict__ knnIdx,
    const float* __restrict__ centers, const float* __restrict__ w1,
    const float* __restrict__ b1, const float* __restrict__ g1,
    const float* __restrict__ be1, const float* __restrict__ m1,
    const float* __restrict__ v1, const float* __restrict__ w2,
    const float* __restrict__ b2, const float* __restrict__ w3,
    const float* __restrict__ b3, const float* __restrict__ g2,
    const float* __restrict__ be2, const float* __restrict__ m2,
    const float* __restrict__ v2, const float* __restrict__ w4,
    const float* __restrict__ b4, float* __restrict__ tokens) {
  extern __shared__ float smem[];
  float* sFeat = smem;                       // [32][4]
  float* sH1   = smem + 128;                 // [32][132]  (128 + pad 4)
  float* sX3   = smem + 128 + 32 * 132;      // [32][516]  (512 + pad 4)
  float* sH3   = sX3 + 32 * 516;             // [32][516]
  float* sH4   = sX3;                        // alias: [32][388] (384 + pad 4)

  const int gg       = blockIdx.x;
  const int b        = gg / NUM_G;
  const int tid      = threadIdx.x;
  const int lane     = tid & 31;
  const int wv       = tid >> 5;
  const int laneN    = lane & 15;
  const int laneHalf = lane >> 4;

  const v8f vzero = {0.f, 0.f, 0.f, 0.f, 0.f, 0.f, 0.f, 0.f};

  // ---- Stage A: gather neighborhood, build (dx,dy,dz,|d|) features ----
  if (tid < GK) {
    int idx = knnIdx[(size_t)gg * GK + tid];
    const float* p = points + ((size_t)b * NUM_N + idx) * 3;
    const float* c = centers + (size_t)gg * 3;
    float dx = p[0] - c[0], dy = p[1] - c[1], dz = p[2] - c[2];
    float nr = sqrtf(dx * dx + dy * dy + dz * dz);
    sFeat[tid * 4 + 0] = dx;
    sFeat[tid * 4 + 1] = dy;
    sFeat[tid * 4 + 2] = dz;
    sFeat[tid * 4 + 3] = nr;
  }
  __syncthreads();

  // ---- Stage B: conv1(4->128) + bn1 + relu -> sH1 (1Mx2N per wave) ----
  {
    int m0 = wv * 16;
    v8f a0 = vzero, a1 = vzero;
    gemm_1x2<4, 4>(w1, 4, sFeat, m0, laneN, laneHalf, a0, a1);
    auto op = [&](float x, int m) {
      float inv = g1[m] * rsqrtf(v1[m] + 1e-5f);
      float v   = (x + b1[m]) * inv + (be1[m] - m1[m] * inv);
      return fmaxf(v, 0.f);
    };
    store_tile(sH1, 132, a0, m0, 0,  laneN, laneHalf, op);
    store_tile(sH1, 132, a1, m0, 16, laneN, laneHalf, op);
  }
  __syncthreads();

  // ---- Stage C: conv2(128->256) + b2 -> sX3[:, 256+m] (2Mx2N per wave) ----
  {
    int m0 = wv * 32;
    v8f a00 = vzero, a01 = vzero, a10 = vzero, a11 = vzero;
    gemm_2x2<C1, 132>(w2, C1, sH1, m0, laneN, laneHalf, a00, a01, a10, a11);
    auto op = [&](float x, int m) { return x + b2[m]; };
    store_tile(sX3 + 256, 516, a00, m0,      0,  laneN, laneHalf, op);
    store_tile(sX3 + 256, 516, a01, m0,      16, laneN, laneHalf, op);
    store_tile(sX3 + 256, 516, a10, m0 + 16, 0,  laneN, laneHalf, op);
    store_tile(sX3 + 256, 516, a11, m0 + 16, 16, laneN, laneHalf, op);
  }
  __syncthreads();

  // ---- Stage D: global max over K, broadcast into sX3[:, 0..255] ----
  {
    int   m  = tid;  // exactly 256 channels
    float gm = -3.4e38f;
    for (int n = 0; n < GK; ++n) gm = fmaxf(gm, sX3[n * 516 + 256 + m]);
    for (int n = 0; n < GK; ++n) sX3[n * 516 + m] = gm;
  }
  __syncthreads();

  // ---- Stage E: conv3(512->512) + bn2 + relu -> sH3 (2 x (2Mx2N)) ----
  {
    auto op = [&](float x, int m) {
      float inv = g2[m] * rsqrtf(v2[m] + 1e-5f);
      float v   = (x + b3[m]) * inv + (be2[m] - m2[m] * inv);
      return fmaxf(v, 0.f);
    };
#pragma unroll
    for (int j = 0; j < 2; ++j) {
      int m0  = wv * 64 + j * 32;
      v8f a00 = vzero, a01 = vzero, a10 = vzero, a11 = vzero;
      gemm_2x2<C3, 516>(w3, C3, sX3, m0, laneN, laneHalf, a00, a01, a10, a11);
      store_tile(sH3, 516, a00, m0,      0,  laneN, laneHalf, op);
      store_tile(sH3, 516, a01, m0,      16, laneN, laneHalf, op);
      store_tile(sH3, 516, a10, m0 + 16, 0,  laneN, laneHalf, op);
      store_tile(sH3, 516, a11, m0 + 16, 16, laneN, laneHalf, op);
    }
  }
  __syncthreads();

  // ---- Stage F: conv4(512->384) + b4 -> sH4 (2Mx2N + 1Mx2N per wave) ----
  {
    auto op = [&](float x, int m) { return x + b4[m]; };
    {
      int m0  = wv * 48;
      v8f a00 = vzero, a01 = vzero, a10 = vzero, a11 = vzero;
      gemm_2x2<C3, 516>(w4, C3, sH3, m0, laneN, laneHalf, a00, a01, a10, a11);
      store_tile(sH4, 388, a00, m0,      0,  laneN, laneHalf, op);
      store_tile(sH4, 388, a01, m0,      16, laneN, laneHalf, op);
      store_tile(sH4, 388, a10, m0 + 16, 0,  laneN, laneHalf, op);
      store_tile(sH4, 388, a11, m0 + 16, 16, laneN, laneHalf, op);
    }
    {
      int m0 = wv * 48 + 32;
      v8f a0 = vzero, a1 = vzero;
      gemm_1x2<C3, 516>(w4, C3, sH3, m0, laneN, laneHalf, a0, a1);
      store_tile(sH4, 388, a0, m0, 0,  laneN, laneHalf, op);
      store_tile(sH4, 388, a1, m0, 16, laneN, laneHalf, op);
    }
  }
  __syncthreads();

  // ---- Stage G: token = max over K ----
  for (int m = tid; m < CT; m += 256) {
    float mx = -3.4e38f;
    for (int n = 0; n < GK; ++n) mx = fmaxf(mx, sH4[n * 388 + m]);
    tokens[(size_t)gg * CT + m] = mx;
  }
}

// ---------------------------------------------------------------------------
extern "C" void kernel_launch(void* const* d_in, const int* in_sizes, int n_in,
                              void* d_out, int out_size, void* d_ws,
                              size_t ws_size, hipStream_t stream) {
  (void)in_sizes; (void)n_in; (void)out_size; (void)ws_size;
  const float* points = (const float*)d_in[0];
  const float* w1  = (const float*)d_in[1];
  const float* b1  = (const float*)d_in[2];
  const float* g1  = (const float*)d_in[3];
  const float* be1 = (const float*)d_in[4];
  const float* m1  = (const float*)d_in[5];
  const float* v1  = (const float*)d_in[6];
  const float* w2  = (const float*)d_in[7];
  const float* b2  = (const float*)d_in[8];
  const float* w3  = (const float*)d_in[9];
  const float* b3  = (const float*)d_in[10];
  const float* g2  = (const float*)d_in[11];
  const float* be2 = (const float*)d_in[12];
  const float* m2  = (const float*)d_in[13];
  const float* v2  = (const float*)d_in[14];
  const float* w4  = (const float*)d_in[15];
  const float* b4  = (const float*)d_in[16];

  float* tokens  = (float*)d_out;             // (8,256,384)
  float* centers = tokens + TOK_TOTAL;        // (8,256,3) tuple tail
  int*   knnIdx  = (int*)d_ws;                // 8*256*32 ints = 256 KB

  fps_kernel<<<NUM_B, 1024, 0, stream>>>(points, centers);
  knn_kernel<<<NUM_B * NUM_G, 256, 0, stream>>>(points, centers, knnIdx);

  size_t ldsBytes = (size_t)(128 + 32 * 132 + 32 * 516 + 32 * 516) * sizeof(float);
  encoder_kernel<<<NUM_B * NUM_G, 256, ldsBytes, stream>>>(
      points, knnIdx, centers, w1, b1, g1, be1, m1, v1, w2, b2, w3, b3, g2,
      be2, m2, v2, w4, b4, tokens);
}